// AttentionHead_6743098655171
// MI455X (gfx1250) — compile-verified
//
#include <hip/hip_runtime.h>

typedef float v2f __attribute__((ext_vector_type(2)));
typedef float v8f __attribute__((ext_vector_type(8)));

#define N_CTX   4096
#define D_MODEL 1024
#define D_HEAD  64

// D = A(16x4 f32) * B(4x16 f32) + C(16x16 f32), full fp32 matrix op (CDNA5)
__device__ __forceinline__ v8f wmma_f32(v2f a, v2f b, v8f c) {
    return __builtin_amdgcn_wmma_f32_16x16x4_f32(
        /*neg_a=*/false, a, /*neg_b=*/false, b,
        /*c_mod=*/(short)0, c, /*reuse_a=*/false, /*reuse_b=*/false);
}

// ---------------------------------------------------------------------------
// Kernel A: q = x@Wq, k = x@WkT^T, t = x@Wo   (three [4096,1024]x[1024,64])
// grid = (256 row-blocks, 4 col-blocks), 1 wave per block, A tile shared.
// ---------------------------------------------------------------------------
__global__ __launch_bounds__(32)
void qkt_kernel(const float* __restrict__ x,  const float* __restrict__ Wq,
                const float* __restrict__ WkT, const float* __restrict__ Wo,
                float* __restrict__ q, float* __restrict__ k, float* __restrict__ t)
{
    const int lane = threadIdx.x;
    const int row0 = blockIdx.x * 16;
    const int col0 = blockIdx.y * 16;
    const int n    = lane & 15;        // matrix column (B/C/D) or A row
    const int hi   = lane >> 4;        // lane half
    const int koff = hi * 2;           // K sub-offset within 4-wide step

    v8f cq = {}, ck = {}, ct = {};
    const float* xrow = x + (size_t)(row0 + n) * D_MODEL;

    #pragma unroll 8
    for (int kk = 0; kk < D_MODEL; kk += 4) {
        v2f a;                                    // A: x[row0+n][kk+koff .. +1]
        a.x = xrow[kk + koff];
        a.y = xrow[kk + koff + 1];

        v2f bq, bk, bt;                           // B: rows kk+koff, kk+koff+1
        bq.x = Wq [(size_t)(kk + koff)     * D_HEAD  + col0 + n];
        bq.y = Wq [(size_t)(kk + koff + 1) * D_HEAD  + col0 + n];
        bk.x = WkT[(size_t)(col0 + n) * D_MODEL + kk + koff];       // k = x@WkT^T
        bk.y = WkT[(size_t)(col0 + n) * D_MODEL + kk + koff + 1];
        bt.x = Wo [(size_t)(kk + koff)     * D_HEAD  + col0 + n];
        bt.y = Wo [(size_t)(kk + koff + 1) * D_HEAD  + col0 + n];

        cq = wmma_f32(a, bq, cq);
        ck = wmma_f32(a, bk, ck);
        ct = wmma_f32(a, bt, ct);
    }

    #pragma unroll
    for (int v = 0; v < 8; ++v) {
        const int r = row0 + v + 8 * hi;
        q[(size_t)r * D_HEAD + col0 + n] = cq[v];
        k[(size_t)r * D_HEAD + col0 + n] = ck[v];
        t[(size_t)r * D_HEAD + col0 + n] = ct[v];
    }
}

// ---------------------------------------------------------------------------
// Kernel B: flash attention.  u = softmax_causal(q@k^T) @ t   ([4096,64])
// grid = 256 query blocks, 1 wave each; online softmax; P re-laid via LDS.
// ---------------------------------------------------------------------------
__global__ __launch_bounds__(32)
void flash_kernel(const float* __restrict__ q, const float* __restrict__ k,
                  const float* __restrict__ t, float* __restrict__ u)
{
    __shared__ float sP[16 * 17];      // padded 16x16 staging tile
    const int lane = threadIdx.x;
    const int i0   = blockIdx.x * 16;
    const int n    = lane & 15;
    const int hi   = lane >> 4;
    const int koff = hi * 2;

    // Q block resident in registers as 16 A-fragments (K = 64)
    v2f aq[16];
    #pragma unroll
    for (int f = 0; f < 16; ++f) {
        aq[f].x = q[(size_t)(i0 + n) * D_HEAD + 4 * f + koff];
        aq[f].y = q[(size_t)(i0 + n) * D_HEAD + 4 * f + koff + 1];
    }

    v8f o0 = {}, o1 = {}, o2 = {}, o3 = {};
    float mst[8], lst[8];
    #pragma unroll
    for (int v = 0; v < 8; ++v) { mst[v] = -__builtin_inff(); lst[v] = 0.f; }

    for (int j0 = 0; j0 <= i0; j0 += 16) {
        // ---- S = Q @ K^T (16x16 tile, 16 WMMA steps over K=64) ----
        v8f s = {};
        #pragma unroll
        for (int f = 0; f < 16; ++f) {
            v2f b;  // B[kk][n] = k[(j0+n)*64 + 4f+kk]
            b.x = k[(size_t)(j0 + n) * D_HEAD + 4 * f + koff];
            b.y = k[(size_t)(j0 + n) * D_HEAD + 4 * f + koff + 1];
            s = wmma_f32(aq[f], b, s);
        }

        if (j0 == i0) {                 // causal mask on diagonal block
            #pragma unroll
            for (int v = 0; v < 8; ++v) {
                const int row = v + 8 * hi;
                if (n > row) s[v] = -__builtin_inff();
            }
        }

        // ---- online softmax (row = v + 8*hi lives in one 16-lane half) ----
        #pragma unroll
        for (int v = 0; v < 8; ++v) {
            float rmax = s[v];
            #pragma unroll
            for (int off = 1; off < 16; off <<= 1)
                rmax = fmaxf(rmax, __shfl_xor(rmax, off, 32));
            const float mnew  = fmaxf(mst[v], rmax);
            const float alpha = __expf(mst[v] - mnew);
            mst[v] = mnew;
            const float p = __expf(s[v] - mnew);
            s[v] = p;
            float rsum = p;
            #pragma unroll
            for (int off = 1; off < 16; off <<= 1)
                rsum += __shfl_xor(rsum, off, 32);
            lst[v] = lst[v] * alpha + rsum;
            o0[v] *= alpha; o1[v] *= alpha; o2[v] *= alpha; o3[v] *= alpha;
        }

        // ---- P (C-layout) -> LDS -> A-layout fragments ----
        #pragma unroll
        for (int v = 0; v < 8; ++v)
            sP[(v + 8 * hi) * 17 + n] = s[v];
        __syncthreads();

        // ---- O += P @ T_block (K=16: 4 steps; N=64: 4 tiles) ----
        #pragma unroll
        for (int f = 0; f < 4; ++f) {
            v2f ap;
            ap.x = sP[n * 17 + 4 * f + koff];
            ap.y = sP[n * 17 + 4 * f + koff + 1];
            const float* tb = t + (size_t)(j0 + 4 * f + koff) * D_HEAD;
            v2f b0, b1, b2, b3;
            b0.x = tb[ 0 + n]; b0.y = tb[D_HEAD +  0 + n];
            b1.x = tb[16 + n]; b1.y = tb[D_HEAD + 16 + n];
            b2.x = tb[32 + n]; b2.y = tb[D_HEAD + 32 + n];
            b3.x = tb[48 + n]; b3.y = tb[D_HEAD + 48 + n];
            o0 = wmma_f32(ap, b0, o0);
            o1 = wmma_f32(ap, b1, o1);
            o2 = wmma_f32(ap, b2, o2);
            o3 = wmma_f32(ap, b3, o3);
        }
        __syncthreads();
    }

    #pragma unroll
    for (int v = 0; v < 8; ++v) {
        const int r = i0 + v + 8 * hi;
        const float inv = 1.f / lst[v];
        u[(size_t)r * D_HEAD +  0 + n] = o0[v] * inv;
        u[(size_t)r * D_HEAD + 16 + n] = o1[v] * inv;
        u[(size_t)r * D_HEAD + 32 + n] = o2[v] * inv;
        u[(size_t)r * D_HEAD + 48 + n] = o3[v] * inv;
    }
}

// ---------------------------------------------------------------------------
// Kernel C: out = u @ WvT   ([4096,64]x[64,1024] -> [4096,1024])
// grid = (256, 64) tiles, 1 wave per 16x16 tile, K=64 -> 16 WMMA steps.
// ---------------------------------------------------------------------------
__global__ __launch_bounds__(32)
void proj_kernel(const float* __restrict__ u, const float* __restrict__ WvT,
                 float* __restrict__ out)
{
    const int lane = threadIdx.x;
    const int row0 = blockIdx.x * 16;
    const int col0 = blockIdx.y * 16;
    const int n    = lane & 15;
    const int hi   = lane >> 4;
    const int koff = hi * 2;

    v8f c = {};
    #pragma unroll
    for (int f = 0; f < 16; ++f) {
        v2f a, b;
        a.x = u[(size_t)(row0 + n) * D_HEAD + 4 * f + koff];
        a.y = u[(size_t)(row0 + n) * D_HEAD + 4 * f + koff + 1];
        b.x = WvT[(size_t)(4 * f + koff)     * D_MODEL + col0 + n];
        b.y = WvT[(size_t)(4 * f + koff + 1) * D_MODEL + col0 + n];
        c = wmma_f32(a, b, c);
    }
    #pragma unroll
    for (int v = 0; v < 8; ++v)
        out[(size_t)(row0 + v + 8 * hi) * D_MODEL + col0 + n] = c[v];
}

// ---------------------------------------------------------------------------
extern "C" void kernel_launch(void* const* d_in, const int* in_sizes, int n_in,
                              void* d_out, int out_size, void* d_ws, size_t ws_size,
                              hipStream_t stream) {
    const float* x   = (const float*)d_in[0];   // [4096,1024]
    const float* Wq  = (const float*)d_in[1];   // [1024,64]
    const float* WkT = (const float*)d_in[2];   // [64,1024]
    const float* Wo  = (const float*)d_in[3];   // [1024,64]
    const float* WvT = (const float*)d_in[4];   // [64,1024]
    float* out = (float*)d_out;                 // [4096,1024]

    const size_t SZ = (size_t)N_CTX * D_HEAD;   // 256K floats = 1MB
    float* q = (float*)d_ws;
    float* k = q + SZ;
    float* t = k + SZ;
    float* u = t + SZ;                          // total 4MB scratch

    qkt_kernel<<<dim3(N_CTX / 16, D_HEAD / 16), 32, 0, stream>>>(x, Wq, WkT, Wo, q, k, t);
    flash_kernel<<<dim3(N_CTX / 16), 32, 0, stream>>>(q, k, t, u);
    proj_kernel<<<dim3(N_CTX / 16, D_MODEL / 16), 32, 0, stream>>>(u, WvT, out);
}